// GaussianIndexedRegressionLayer_13804024889975
// MI455X (gfx1250) — compile-verified
//
#include <hip/hip_runtime.h>
#include <math.h>

typedef __attribute__((ext_vector_type(2))) float v2f;
typedef __attribute__((ext_vector_type(8))) float v8f;

#define TINYF 1.1754943508222875e-38f

// ---------------------------------------------------------------------------
// Kernel 1: transpose W (D x 2R, row-major) -> WT (2R x D) so each gathered
// "column" becomes a contiguous 4KB row (coalesced b64 streaming in kernel 2).
// ---------------------------------------------------------------------------
__global__ void wt_transpose(const float* __restrict__ W, float* __restrict__ WT,
                             int Dd, int R2) {
  __shared__ float tile[32][33];
  const int c0 = blockIdx.x * 32;   // along 2R dimension
  const int d0 = blockIdx.y * 32;   // along D dimension
  const int tx = threadIdx.x;       // 0..31
  const int ty = threadIdx.y;       // 0..7
#pragma unroll
  for (int i = 0; i < 32; i += 8)
    tile[ty + i][tx] = W[(long)(d0 + ty + i) * R2 + (c0 + tx)];
  __syncthreads();
#pragma unroll
  for (int i = 0; i < 32; i += 8)
    WT[(long)(c0 + ty + i) * Dd + (d0 + tx)] = tile[tx][ty + i];
}

// ---------------------------------------------------------------------------
// Kernel 2: one wave per (b,s) row.
//   A (16x4 f32)  = 16 gathered W rows (mean: 2*idx, std: 2*idx+1)
//   B (4x16 f32)  = X row broadcast into all 16 columns (LDS ds_load_b64)
//   D = A x B + C accumulates dot(W_row_m, x) in every column; read col 0
//   from lanes 0 (M=0..7) and 16 (M=8..15) per the CDNA5 C/D VGPR layout.
//   X row is staged into LDS with GLOBAL_LOAD_ASYNC_TO_LDS_B128 (ASYNCcnt).
// ---------------------------------------------------------------------------
template <bool TR>
__global__ void gauss_gather(const float* __restrict__ X,
                             const int* __restrict__ idx,
                             const float* __restrict__ Wsrc,  // TR ? WT : W
                             const float* __restrict__ bias,
                             float* __restrict__ mean_out,
                             float* __restrict__ std_out) {
  constexpr int D = 1024, P = 16, R2 = 16384;
  __shared__ float xs[8][D];  // 8 waves * 4KB = 32KB LDS (offset 0 in LDS)
  const int wave = threadIdx.x >> 5;
  const int lane = threadIdx.x & 31;
  const long row = (long)blockIdx.x * 8 + wave;

  // Stage this wave's X row into LDS via the CDNA5 async DMA path:
  // one b128 per lane per step, tracked by ASYNCcnt (no VGPR round trip).
  const float* xr = X + row * D;
#pragma unroll
  for (int i = 0; i < D / 128; ++i) {
    const int e = (i * 32 + lane) * 4;
    const unsigned ldsoff = (unsigned)((wave * D + e) * sizeof(float));
    const unsigned long long ga = (unsigned long long)(xr + e);
    asm volatile("global_load_async_to_lds_b128 %0, %1, off"
                 :: "v"(ldsoff), "v"(ga)
                 : "memory");
  }
  asm volatile("s_wait_asynccnt 0x0" ::: "memory");
  __syncthreads();

  const int p    = lane & 15;  // which of the 16 gathered indices (M row)
  const int half = lane >> 4;  // K-half per the 32-bit A/B layouts
  const int gi   = idx[row * P + p];

  long moff, soff;
  if (TR) {
    moff = (long)(2 * gi) * D;  // contiguous 4KB row in WT
    soff = moff + D;
  } else {
    moff = 2 * gi;              // strided column in original W
    soff = moff + 1;
  }

  v8f accm = {};
  v8f accs = {};

#pragma unroll 8
  for (int k = 0; k < D; k += 4) {
    const int kk = k + half * 2;  // lanes 0-15: K=k..k+1, lanes 16-31: K=k+2..k+3
    v2f am, as_, bm;
    if (TR) {
      am  = *(const v2f*)(Wsrc + moff + kk);   // global_load_b64, streaming
      as_ = *(const v2f*)(Wsrc + soff + kk);
    } else {
      am.x  = Wsrc[(long)kk * R2 + moff];
      am.y  = Wsrc[(long)(kk + 1) * R2 + moff];
      as_.x = Wsrc[(long)kk * R2 + soff];
      as_.y = Wsrc[(long)(kk + 1) * R2 + soff];
    }
    bm = *(const v2f*)(&xs[wave][kk]);         // ds_load_b64, broadcast per half

    accm = __builtin_amdgcn_wmma_f32_16x16x4_f32(false, am, false, bm,
                                                 (short)0, accm, false, false);
    accs = __builtin_amdgcn_wmma_f32_16x16x4_f32(false, as_, false, bm,
                                                 (short)0, accs, false, false);
  }

  // D-matrix column 0 lives in lane 0 (M=0..7) and lane 16 (M=8..15).
  if (p == 0) {
    const int mbase = half * 8;
#pragma unroll
    for (int j = 0; j < 8; ++j) {
      const int m = mbase + j;
      const int g = idx[row * P + m];
      const float mu = accm[j] + bias[2 * g];
      const float z  = accs[j] + bias[2 * g + 1];
      // elu(z) + 1 + TINY
      const float sd = (z > 0.f) ? (z + 1.f + TINYF) : (__expf(z) + TINYF);
      mean_out[row * P + m] = mu;
      std_out[row * P + m]  = sd;
    }
  }
}

// ---------------------------------------------------------------------------
extern "C" void kernel_launch(void* const* d_in, const int* in_sizes, int n_in,
                              void* d_out, int out_size, void* d_ws,
                              size_t ws_size, hipStream_t stream) {
  const float* X  = (const float*)d_in[0];  // (B,S,D) f32
  const int* idx  = (const int*)d_in[1];    // (B,S,P) i32
  const float* W  = (const float*)d_in[2];  // (D,2R)  f32
  const float* bb = (const float*)d_in[3];  // (2R,)   f32

  const int D  = 1024;
  const int P  = 16;
  const int R2 = 16384;
  const int BS = in_sizes[0] / D;  // 8192 rows total

  float* mean_out = (float*)d_out;
  float* std_out  = mean_out + (long)BS * P;

  const size_t wt_bytes = (size_t)R2 * D * sizeof(float);
  if (ws_size >= wt_bytes) {
    float* WT = (float*)d_ws;
    wt_transpose<<<dim3(R2 / 32, D / 32), dim3(32, 8), 0, stream>>>(W, WT, D, R2);
    gauss_gather<true><<<BS / 8, 256, 0, stream>>>(X, idx, WT, bb, mean_out, std_out);
  } else {
    gauss_gather<false><<<BS / 8, 256, 0, stream>>>(X, idx, W, bb, mean_out, std_out);
  }
}